// RoIAlign_4337916968972
// MI455X (gfx1250) — compile-verified
//
#include <hip/hip_runtime.h>
#include <stdint.h>

typedef float v2f __attribute__((ext_vector_type(2)));
typedef float v8f __attribute__((ext_vector_type(8)));

#define PH_    7
#define PW_    7
#define NBINS  49
#define SCALE_ 0.25f
#define C_     256
#define H_     200
#define W_     200
#define NWAVE  8
#define NTHR   (NWAVE * 32)

// One block per ROI. 8 waves; wave w owns bins {w, w+8, ...} (7 or 6 bins).
// Phase 1: precompute per-(bin,sample) gather BYTE offsets + masked bilinear weights in LDS.
// Phase 2: per 16-channel tile, accumulate out[16ch x 16bin] with V_WMMA_F32_16X16X4_F32.
//   A (16x4 f32): lanes 0-15 hold K={0,1}, lanes 16-31 hold K={2,3} -> 2 gathers/lane/chunk.
//   B (4x16 f32): one-hot weight column (validity mask + 1/4 averaging folded into weights).
// Image base is readfirstlane'd into SGPRs so gathers are SADDR-form with u32 voffsets.
__global__ __launch_bounds__(NTHR) void roialign_wmma_f32(
    const float* __restrict__ input, const float* __restrict__ rois,
    float* __restrict__ out)
{
    __shared__ int   s_off[NBINS * 16];   // byte offsets within one channel plane
    __shared__ float s_wt [NBINS * 16];

    const int r = blockIdx.x;
    const float* roi = rois + (size_t)r * 5;
    const int   bimg = __builtin_amdgcn_readfirstlane((int)roi[0]); // uniform -> SGPR base
    const float sx = roi[1] * SCALE_ - 0.5f;
    const float sy = roi[2] * SCALE_ - 0.5f;
    const float bw = (roi[3] - roi[1]) * SCALE_ * (1.0f / PW_);
    const float bh = (roi[4] - roi[2]) * SCALE_ * (1.0f / PH_);

    // ---- Phase 1: coordinate/weight precompute (once per ROI) ----
    for (int idx = threadIdx.x; idx < NBINS * 16; idx += NTHR) {
        const int bin    = idx >> 4;        // 0..48
        const int s      = idx & 15;        // sample slot
        const int sub    = s >> 2;          // 0..3 : (sy_sub, sx_sub)
        const int corner = s & 3;           // 0..3 : (y0x0, y0x1, y1x0, y1x1)
        const int py = bin / PW_;
        const int px = bin - py * PW_;
        const float yy = sy + ((float)py + ((float)(sub >> 1) + 0.5f) * 0.5f) * bh;
        const float xx = sx + ((float)px + ((float)(sub & 1)  + 0.5f) * 0.5f) * bw;
        const bool valid = (yy >= -1.0f) && (yy <= (float)H_) &&
                           (xx >= -1.0f) && (xx <= (float)W_);
        const float yc = fminf(fmaxf(yy, 0.0f), (float)(H_ - 1));
        const float xc = fminf(fmaxf(xx, 0.0f), (float)(W_ - 1));
        const int y0 = (int)floorf(yc);
        const int x0 = (int)floorf(xc);
        const int y1 = min(y0 + 1, H_ - 1);
        const int x1 = min(x0 + 1, W_ - 1);
        const float ly = yc - (float)y0, lx = xc - (float)x0;
        const float hy = 1.0f - ly,      hx = 1.0f - lx;
        const int   ys = (corner & 2) ? y1 : y0;
        const int   xs = (corner & 1) ? x1 : x0;
        const float wy = (corner & 2) ? ly : hy;
        const float wx = (corner & 1) ? lx : hx;
        s_off[idx] = (ys * W_ + xs) << 2;                   // byte offset, always in-bounds
        s_wt [idx] = valid ? (0.25f * wy * wx) : 0.0f;      // masks invalid + SRxSR average
    }
    __syncthreads();

    const int wave  = threadIdx.x >> 5;
    const int lane  = threadIdx.x & 31;
    const int nhalf = lane >> 4;          // 0: lanes 0-15, 1: lanes 16-31
    const int l15   = lane & 15;
    const int khalf = nhalf << 1;         // K offset per A/B lane half: 0 or 2
    const int nb    = (wave == 0) ? 7 : 6;

    // Scalar (SGPR) image base; per-lane channel offset stays a u32 byte offset.
    const char* imgb = (const char*)input +
                       (size_t)bimg * ((size_t)C_ * H_ * W_ * sizeof(float));

    for (int t = 0; t < 16; ++t) {
        const uint32_t cb = (uint32_t)(t * 16 + l15) * (uint32_t)(H_ * W_ * sizeof(float));
        v8f accA = {0.f, 0.f, 0.f, 0.f, 0.f, 0.f, 0.f, 0.f};
        v8f accB = {0.f, 0.f, 0.f, 0.f, 0.f, 0.f, 0.f, 0.f};

        for (int bl = 0; bl < nb; ++bl) {
            const int bin  = wave + (bl << 3);
            const int base = (bin << 4) + khalf;

            // u32 voffsets -> SADDR-form global_load_b32 gathers (batched).
            const uint32_t o0x = cb + (uint32_t)s_off[base +  0];
            const uint32_t o0y = cb + (uint32_t)s_off[base +  1];
            const uint32_t o1x = cb + (uint32_t)s_off[base +  4];
            const uint32_t o1y = cb + (uint32_t)s_off[base +  5];
            const uint32_t o2x = cb + (uint32_t)s_off[base +  8];
            const uint32_t o2y = cb + (uint32_t)s_off[base +  9];
            const uint32_t o3x = cb + (uint32_t)s_off[base + 12];
            const uint32_t o3y = cb + (uint32_t)s_off[base + 13];
            v2f a0 = { *(const float*)(imgb + o0x), *(const float*)(imgb + o0y) };
            v2f a1 = { *(const float*)(imgb + o1x), *(const float*)(imgb + o1y) };
            v2f a2 = { *(const float*)(imgb + o2x), *(const float*)(imgb + o2y) };
            v2f a3 = { *(const float*)(imgb + o3x), *(const float*)(imgb + o3y) };

            // One-hot weight column: only column l15 == bl is non-zero.
            const float sel = (l15 == bl) ? 1.0f : 0.0f;
            v2f b0 = { sel * s_wt[base +  0], sel * s_wt[base +  1] };
            v2f b1 = { sel * s_wt[base +  4], sel * s_wt[base +  5] };
            v2f b2 = { sel * s_wt[base +  8], sel * s_wt[base +  9] };
            v2f b3 = { sel * s_wt[base + 12], sel * s_wt[base + 13] };

            // Two accumulation chains -> 2-deep ILP on the matrix pipe.
            accA = __builtin_amdgcn_wmma_f32_16x16x4_f32(false, a0, false, b0,
                                                         (short)0, accA, false, false);
            accB = __builtin_amdgcn_wmma_f32_16x16x4_f32(false, a1, false, b1,
                                                         (short)0, accB, false, false);
            accA = __builtin_amdgcn_wmma_f32_16x16x4_f32(false, a2, false, b2,
                                                         (short)0, accA, false, false);
            accB = __builtin_amdgcn_wmma_f32_16x16x4_f32(false, a3, false, b3,
                                                         (short)0, accB, false, false);
        }

        const v8f acc = accA + accB;

        // D layout: VGPR j -> M = j + nhalf*8, N = l15. Column n maps to bin wave + 8n.
        if (l15 < nb) {
            const int bin = wave + (l15 << 3);
            size_t obase = ((size_t)r * C_ + (size_t)(t * 16 + nhalf * 8)) * NBINS + bin;
            #pragma unroll
            for (int j = 0; j < 8; ++j)
                out[obase + (size_t)j * NBINS] = acc[j];
        }
    }
}

extern "C" void kernel_launch(void* const* d_in, const int* in_sizes, int n_in,
                              void* d_out, int out_size, void* d_ws, size_t ws_size,
                              hipStream_t stream) {
    const float* input = (const float*)d_in[0];
    const float* rois  = (const float*)d_in[1];
    float* out = (float*)d_out;
    const int n_rois = in_sizes[1] / 5;
    if (n_rois <= 0) return;
    roialign_wmma_f32<<<n_rois, NTHR, 0, stream>>>(input, rois, out);
}